// GAE_19241453486393
// MI455X (gfx1250) — compile-verified
//
#include <hip/hip_runtime.h>
#include <hip/hip_bf16.h>

// GAE backward scan, B rows x T timesteps, row-major [B, T].
// One wave32 per row. Inputs are staged into LDS with gfx1250
// global_load_async_to_lds_* (ASYNCcnt-tracked DMA), double-buffered at
// distance 2. Per k-step: 128 contiguous t-values, lane owns 4 (16B async
// copy per lane per array -> fully coalesced 512B per wave). The serial
// recurrence is resolved with an affine-map suffix scan across lanes
// (ds_bpermute); a scalar carry chains the 16 k-steps.

#define T_LEN  2048
#define GAMMA  0.99f
#define GLAM   (0.99f * 0.95f)   // gamma * lambda
#define WPB    8                 // waves (rows) per block

// Truncate a generic pointer to a shared-memory object down to the 32-bit
// LDS byte address (generic LDS pointers carry the DS offset in addr[31:0]).
__device__ __forceinline__ unsigned lds_addr32(const void* p) {
    return (unsigned)(unsigned long long)p;
}

// One async stage: copy 128 floats from each f32 array + 128 bool bytes
// into this wave's LDS buffer. 4 async ops -> ASYNCcnt += 4.
__device__ __forceinline__ void issue_stage(
    const float* rbase, const float* vbase, const float* nbase,
    const unsigned char* tbase,
    long long elem_base,              // row_base + k*128 (element index)
    int lane,
    float* lds_r, float* lds_v, float* lds_n, unsigned* lds_t)
{
    const unsigned go_f = (unsigned)((elem_base + lane * 4) * 4); // byte off (f32)
    const unsigned go_b = (unsigned)( elem_base + lane * 4);      // byte off (bool)
    const unsigned lr = lds_addr32(lds_r + lane * 4);
    const unsigned lv = lds_addr32(lds_v + lane * 4);
    const unsigned ln = lds_addr32(lds_n + lane * 4);
    const unsigned lt = lds_addr32(lds_t + lane);
    const unsigned long long sr = (unsigned long long)rbase;
    const unsigned long long sv = (unsigned long long)vbase;
    const unsigned long long sn = (unsigned long long)nbase;
    const unsigned long long st = (unsigned long long)tbase;

    asm volatile("global_load_async_to_lds_b128 %0, %1, %2"
                 :: "v"(lr), "v"(go_f), "s"(sr) : "memory");
    asm volatile("global_load_async_to_lds_b128 %0, %1, %2"
                 :: "v"(lv), "v"(go_f), "s"(sv) : "memory");
    asm volatile("global_load_async_to_lds_b128 %0, %1, %2"
                 :: "v"(ln), "v"(go_f), "s"(sn) : "memory");
    asm volatile("global_load_async_to_lds_b32 %0, %1, %2"
                 :: "v"(lt), "v"(go_b), "s"(st) : "memory");
}

__global__ __launch_bounds__(WPB * 32) void gae_wave_scan_async(
    const float*         __restrict__ reward,
    const unsigned char* __restrict__ terminated,   // jnp.bool_ -> 1 byte/elem
    const float*         __restrict__ value,
    const float*         __restrict__ next_value,
    float*               __restrict__ adv,
    float*               __restrict__ ret,
    int B)
{
    __shared__ __align__(16) float    s_r[WPB][2][128];
    __shared__ __align__(16) float    s_v[WPB][2][128];
    __shared__ __align__(16) float    s_n[WPB][2][128];
    __shared__ __align__(16) unsigned s_t[WPB][2][32];

    const int lane = threadIdx.x & 31;
    const int wave = threadIdx.x >> 5;
    const int row  = blockIdx.x * WPB + wave;
    if (row >= B) return;                      // uniform per wave; no barriers used

    const long long base = (long long)row * T_LEN;
    const int KMAX = T_LEN / 128;              // 16 stages

    // Prologue: prefetch stages 15 (buf 1) and 14 (buf 0).
    issue_stage(reward, value, next_value, terminated, base + (KMAX - 1) * 128,
                lane, s_r[wave][1], s_v[wave][1], s_n[wave][1], s_t[wave][1]);
    issue_stage(reward, value, next_value, terminated, base + (KMAX - 2) * 128,
                lane, s_r[wave][0], s_v[wave][0], s_n[wave][0], s_t[wave][0]);

    float carry = 0.0f;                        // gae flowing backward across stages

    #pragma unroll 1
    for (int k = KMAX - 1; k >= 0; --k) {
        const int buf = k & 1;

        // Stage k landed when <= 4 async ops remain (in-order completion);
        // at k==0 nothing younger is in flight, so drain fully.
        if (k > 0) asm volatile("s_wait_asynccnt 0x4" ::: "memory");
        else       asm volatile("s_wait_asynccnt 0x0" ::: "memory");

        const float4 r4 = *(const float4*)&s_r[wave][buf][lane * 4];
        const float4 v4 = *(const float4*)&s_v[wave][buf][lane * 4];
        const float4 n4 = *(const float4*)&s_n[wave][buf][lane * 4];
        const unsigned tw = s_t[wave][buf][lane];

        // Reads must be in VGPRs before the DMA for stage k-2 reuses this buf.
        asm volatile("s_wait_dscnt 0x0" ::: "memory");
        if (k >= 2)
            issue_stage(reward, value, next_value, terminated, base + (k - 2) * 128,
                        lane, s_r[wave][buf], s_v[wave][buf], s_n[wave][buf],
                        s_t[wave][buf]);

        const float nd0 = (tw & 0x000000FFu) ? 0.0f : 1.0f;
        const float nd1 = (tw & 0x0000FF00u) ? 0.0f : 1.0f;
        const float nd2 = (tw & 0x00FF0000u) ? 0.0f : 1.0f;
        const float nd3 = (tw & 0xFF000000u) ? 0.0f : 1.0f;

        // delta_t = r + gamma*nv*nd - v ;  c_t = gamma*lambda*nd
        const float d0 = fmaf(GAMMA * nd0, n4.x, r4.x) - v4.x;
        const float d1 = fmaf(GAMMA * nd1, n4.y, r4.y) - v4.y;
        const float d2 = fmaf(GAMMA * nd2, n4.z, r4.z) - v4.z;
        const float d3 = fmaf(GAMMA * nd3, n4.w, r4.w) - v4.w;
        const float c0 = GLAM * nd0;
        const float c1 = GLAM * nd1;
        const float c2 = GLAM * nd2;
        const float c3 = GLAM * nd3;

        // Local affine map of this lane's 4 elements (backward over t):
        //   g_in (gae at t0+4)  ->  D + C * g_in  (gae at t0)
        float C = c3, D = d3;
        D = fmaf(c2, D, d2);  C *= c2;
        D = fmaf(c1, D, d1);  C *= c1;
        D = fmaf(c0, D, d0);  C *= c0;

        // Inclusive suffix composition across lanes (higher lane = later t):
        //   S_l = M_l o M_{l+1} o ... o M_31
        float sc = C, sd = D;
        #pragma unroll
        for (int off = 1; off < 32; off <<= 1) {
            const float oc = __shfl_down(sc, (unsigned)off, 32);
            const float od = __shfl_down(sd, (unsigned)off, 32);
            if (lane + off < 32) {             // compose: self o fetched
                sd = fmaf(sc, od, sd);
                sc = sc * oc;
            }
        }

        // Incoming gae = (exclusive suffix of lanes above)(carry)
        float ic  = __shfl_down(sc, 1u, 32);
        float idn = __shfl_down(sd, 1u, 32);
        if (lane == 31) { ic = 1.0f; idn = 0.0f; }
        const float g_in = fmaf(ic, carry, idn);

        // Replay this lane's 4 elements with the correct incoming gae.
        const float g3 = fmaf(c3, g_in, d3);
        const float g2 = fmaf(c2, g3,   d2);
        const float g1 = fmaf(c1, g2,   d1);
        const float g0 = fmaf(c0, g1,   d0);

        const long long idx = base + (long long)(k * 128 + lane * 4);
        *(float4*)(adv + idx) = make_float4(g0, g1, g2, g3);
        *(float4*)(ret + idx) = make_float4(g0 + v4.x, g1 + v4.y,
                                            g2 + v4.z, g3 + v4.w);

        // carry across stages: lane 0 holds the full-128-element map
        const float tc = __shfl(sc, 0, 32);
        const float td = __shfl(sd, 0, 32);
        carry = fmaf(tc, carry, td);
    }
}

extern "C" void kernel_launch(void* const* d_in, const int* in_sizes, int n_in,
                              void* d_out, int out_size, void* d_ws, size_t ws_size,
                              hipStream_t stream) {
    const float*         reward     = (const float*)        d_in[0];
    const unsigned char* terminated = (const unsigned char*)d_in[1];
    const float*         value      = (const float*)        d_in[2];
    const float*         next_value = (const float*)        d_in[3];

    const int n = in_sizes[0];          // B * T
    const int B = n / T_LEN;

    float* out_adv = (float*)d_out;                       // [B, T]
    float* out_ret = (float*)d_out + (size_t)B * T_LEN;   // [B, T]

    const int grid = (B + WPB - 1) / WPB;

    gae_wave_scan_async<<<grid, WPB * 32, 0, stream>>>(
        reward, terminated, value, next_value, out_adv, out_ret, B);
}